// GraphEncoder_8297876816596
// MI455X (gfx1250) — compile-verified
//
#include <hip/hip_runtime.h>

#define NN 262144
#define NE 8388608
#define NG 512
#define HID 64
#define ODIM 128

typedef __attribute__((ext_vector_type(2))) float v2f;
typedef __attribute__((ext_vector_type(8))) float v8f;

// ---------------------------------------------------------------- zero scratch
__global__ void gin_zero(float* __restrict__ p, int n) {
  int i = blockIdx.x * blockDim.x + threadIdx.x;
  int stride = gridDim.x * blockDim.x;
  for (; i < n; i += stride) p[i] = 0.0f;
}

// ---------------------------------------------------------------- edge scatter
// agg[dst] += x[src]   (HBM-bound stage: ~0.4GB traffic)
__global__ void gin_edge_scatter(const float4* __restrict__ x4,
                                 const long long* __restrict__ src,
                                 const long long* __restrict__ dst,
                                 float* __restrict__ agg) {
  int e = blockIdx.x * blockDim.x + threadIdx.x;
  if (e >= NE) return;
  long long s = src[e];
  long long d = dst[e];
  float4 v = x4[s];
  float* a = agg + (size_t)d * 4;
  atomicAdd(a + 0, v.x);
  atomicAdd(a + 1, v.y);
  atomicAdd(a + 2, v.z);
  atomicAdd(a + 3, v.w);
}

// ---------------------------------------------------------------- node counts
__global__ void gin_count(const long long* __restrict__ batch,
                          float* __restrict__ counts) {
  int i = blockIdx.x * blockDim.x + threadIdx.x;
  if (i < NN) atomicAdd(&counts[(int)batch[i]], 1.0f);
}

// ---------------------------------------------------------------- fused node MLP + pool
// One wave (32 lanes) processes 16 nodes. GEMM1 is a single K=4 f32 WMMA per
// 16-col tile; GEMM2 chains 16 K=4 WMMAs over HID=64 with 4 accumulators.
// Results are atomically pooled into sums[NG][HID] (lives in L2), so the
// [N,64] intermediate never touches HBM.
__global__ __launch_bounds__(256) void gin_node_mlp_pool(
    const float4* __restrict__ x4, const float4* __restrict__ agg4,
    const long long* __restrict__ batch,
    const float* __restrict__ W1, const float* __restrict__ b1,
    const float* __restrict__ W2, const float* __restrict__ b2,
    float* __restrict__ sums) {
  __shared__ float h1[8][16][HID + 4];   // stride 68 words -> conflict-free b64 reads
  const int tid  = threadIdx.x;
  const int wid  = tid >> 5;
  const int lane = tid & 31;
  const int half = lane >> 4;   // K-half selector for A/B operands
  const int lrow = lane & 15;   // A: row M ; B/C: column within tile
  const int nodeBase = blockIdx.x * 128 + wid * 16;

  // ---- GEMM1: z[16x4] @ W1[4x64] ; z = x + agg
  // A layout (32-bit A 16x4): lanes0-15 hold {K0,K1}, lanes16-31 hold {K2,K3}
  float4 xv = x4[nodeBase + lrow];
  float4 av = agg4[nodeBase + lrow];
  v2f a;
  a[0] = (half == 0) ? (xv.x + av.x) : (xv.z + av.z);
  a[1] = (half == 0) ? (xv.y + av.y) : (xv.w + av.w);

#pragma unroll
  for (int t = 0; t < 4; ++t) {
    const int n = t * 16 + lrow;
    v2f b;
    b[0] = W1[(2 * half + 0) * HID + n];
    b[1] = W1[(2 * half + 1) * HID + n];
    v8f c = {};
    c = __builtin_amdgcn_wmma_f32_16x16x4_f32(false, a, false, b, (short)0, c,
                                              false, false);
    const float bias = b1[n];
#pragma unroll
    for (int v = 0; v < 8; ++v) {
      float hv = c[v] + bias;               // bias + ReLU
      hv = hv > 0.0f ? hv : 0.0f;
      h1[wid][v + 8 * half][n] = hv;        // C layout: reg v -> row v / v+8
    }
  }
  __syncthreads();

  // ---- GEMM2: h1[16x64] @ W2[64x64] + b2, accumulated over 16 K=4 steps
  v8f acc[4] = {{}, {}, {}, {}};
#pragma unroll
  for (int kk = 0; kk < 16; ++kk) {
    const int k0 = 4 * kk + 2 * half;
    v2f a2;
    a2[0] = h1[wid][lrow][k0 + 0];
    a2[1] = h1[wid][lrow][k0 + 1];
#pragma unroll
    for (int t = 0; t < 4; ++t) {
      const int n = t * 16 + lrow;
      v2f b;
      b[0] = W2[(k0 + 0) * HID + n];
      b[1] = W2[(k0 + 1) * HID + n];
      acc[t] = __builtin_amdgcn_wmma_f32_16x16x4_f32(false, a2, false, b,
                                                     (short)0, acc[t], false,
                                                     false);
    }
  }

  // ---- pool: atomic f32 adds into sums[g][n] (128KB, L2-resident)
  int gid[8];
#pragma unroll
  for (int v = 0; v < 8; ++v)
    gid[v] = (int)batch[nodeBase + v + 8 * half];
#pragma unroll
  for (int t = 0; t < 4; ++t) {
    const int n = t * 16 + lrow;
    const float bias = b2[n];
#pragma unroll
    for (int v = 0; v < 8; ++v)
      atomicAdd(&sums[gid[v] * HID + n], acc[t][v] + bias);
  }
}

// ---------------------------------------------------------------- final GEMM
// pooled[512x64] @ Wl[64x128] + bl ; one wave per 16-graph row block.
__global__ __launch_bounds__(256) void gin_final(
    const float* __restrict__ sums, const float* __restrict__ counts,
    const float* __restrict__ Wl, const float* __restrict__ bl,
    float* __restrict__ out) {
  __shared__ float pl[8][16][HID + 4];
  const int tid  = threadIdx.x;
  const int wid  = tid >> 5;
  const int lane = tid & 31;
  const int half = lane >> 4;
  const int lrow = lane & 15;
  const int gBase = (blockIdx.x * 8 + wid) * 16;

  // stage pooled means into LDS
  for (int i = lane; i < 16 * HID; i += 32) {
    const int r = i >> 6;
    const int c = i & 63;
    float cnt = counts[gBase + r];
    cnt = cnt > 1.0f ? cnt : 1.0f;
    pl[wid][r][c] = sums[(gBase + r) * HID + c] / cnt;
  }
  __syncthreads();

  v8f acc[8] = {{}, {}, {}, {}, {}, {}, {}, {}};
#pragma unroll
  for (int kk = 0; kk < 16; ++kk) {
    const int k0 = 4 * kk + 2 * half;
    v2f a;
    a[0] = pl[wid][lrow][k0 + 0];
    a[1] = pl[wid][lrow][k0 + 1];
#pragma unroll
    for (int t = 0; t < 8; ++t) {
      const int n = t * 16 + lrow;
      v2f b;
      b[0] = Wl[(k0 + 0) * ODIM + n];
      b[1] = Wl[(k0 + 1) * ODIM + n];
      acc[t] = __builtin_amdgcn_wmma_f32_16x16x4_f32(false, a, false, b,
                                                     (short)0, acc[t], false,
                                                     false);
    }
  }
#pragma unroll
  for (int t = 0; t < 8; ++t) {
    const int n = t * 16 + lrow;
    const float bias = bl[n];
#pragma unroll
    for (int v = 0; v < 8; ++v)
      out[(gBase + v + 8 * half) * ODIM + n] = acc[t][v] + bias;
  }
}

// ---------------------------------------------------------------- launcher
extern "C" void kernel_launch(void* const* d_in, const int* in_sizes, int n_in,
                              void* d_out, int out_size, void* d_ws,
                              size_t ws_size, hipStream_t stream) {
  const float* x         = (const float*)d_in[0];
  const long long* ei    = (const long long*)d_in[1];  // int64 [2][NE]
  const long long* batch = (const long long*)d_in[2];  // int64 [NN]
  const float* W1 = (const float*)d_in[3];
  const float* b1 = (const float*)d_in[4];
  const float* W2 = (const float*)d_in[5];
  const float* b2 = (const float*)d_in[6];
  const float* Wl = (const float*)d_in[7];
  const float* bl = (const float*)d_in[8];
  float* out = (float*)d_out;

  // workspace layout: agg[NN*4] | sums[NG*HID] | counts[NG]  (~4.3 MB)
  float* agg    = (float*)d_ws;
  float* sums   = agg + (size_t)NN * 4;
  float* counts = sums + (size_t)NG * HID;
  const int nZero = NN * 4 + NG * HID + NG;

  gin_zero<<<2048, 256, 0, stream>>>((float*)d_ws, nZero);
  gin_edge_scatter<<<NE / 256, 256, 0, stream>>>(
      (const float4*)x, ei, ei + NE, agg);
  gin_count<<<NN / 256, 256, 0, stream>>>(batch, counts);
  gin_node_mlp_pool<<<NN / 128, 256, 0, stream>>>(
      (const float4*)x, (const float4*)agg, batch, W1, b1, W2, b2, sums);
  gin_final<<<NG / 128, 256, 0, stream>>>(sums, counts, Wl, bl, out);
}